// Gated_DCMN_Layer_30331059045066
// MI455X (gfx1250) — compile-verified
//
#include <hip/hip_runtime.h>
#include <hip/hip_bf16.h>
#include <math.h>

// ---------------------------------------------------------------------------
// Gated DCMN layer for MI455X (gfx1250).
// Heavy work: 4 streaming passes (2 hops x 2 memories) over the 512MiB f32
// memory tensors with fused online softmax  -> bandwidth bound, ~92us ideal
// at 23.3 TB/s.  Pass kernel: split-L (4 chunks) + 4-rows-per-wave unroll for
// memory-level parallelism, non-temporal loads (no L2 pollution: 512MiB
// stream vs 192MB L2, zero reuse), and prefetch 32KB ahead.
// Small GEMMs (q@E^T, T@F, q@G, q@U) use V_WMMA_F32_16X16X4_F32 (exact f32).
// ---------------------------------------------------------------------------

typedef __attribute__((ext_vector_type(2))) float v2f;
typedef __attribute__((ext_vector_type(4))) float v4f;
typedef __attribute__((ext_vector_type(8))) float v8f;

#define B_SZ 256
#define L_SZ 4096
#define D_SZ 128
#define E_SZ 64
#define NCHUNK 4
#define CHUNK_L (L_SZ / NCHUNK)          // 1024 rows per chunk

__device__ __forceinline__ float dot4(v4f a, v4f b) {
  v4f t = a * b;
  return (t.x + t.y) + (t.z + t.w);
}

// ---------------------------------------------------------------------------
// Generic small GEMM:  C[M,N] (=|+=) A[M,K] @ B   (B addressed by strides so
// we can feed either B[K,N] row-major or B = E^T without a transpose pass).
// One wave per 16x16 C tile; K-loop in steps of 4 via v_wmma_f32_16x16x4_f32.
// VGPR layouts per CDNA5 ISA 7.12.2:
//   A 16x4 : lanes 0-15 -> M=0..15, vgpr{0,1}=K{0,1}; lanes 16-31 -> K{2,3}
//   B 4x16 : lanes 0-15 -> N=0..15, vgpr{0,1}=K{0,1}; lanes 16-31 -> K{2,3}
//   C 16x16: lane lo -> N, vgpr r -> M = r + 8*(lane>=16)
// ---------------------------------------------------------------------------
__global__ __launch_bounds__(32)
void wmma_gemm_f32(const float* __restrict__ A, int lda,
                   const float* __restrict__ Bm, int b_ks, int b_ns,
                   float* __restrict__ C, int ldc,
                   int K, int accumulate) {
  const int lane = threadIdx.x & 31;
  const int lo   = lane & 15;
  const int hi   = lane >> 4;            // 0 or 1
  const int col0 = blockIdx.x * 16;
  const int row0 = blockIdx.y * 16;

  v8f acc = {};
  const float* Arow = A + (size_t)(row0 + lo) * lda;

  for (int k = 0; k < K; k += 4) {
    const int kk = k + 2 * hi;
    v2f a, b;
    a.x = Arow[kk];
    a.y = Arow[kk + 1];
    b.x = Bm[(size_t)kk       * b_ks + (size_t)(col0 + lo) * b_ns];
    b.y = Bm[(size_t)(kk + 1) * b_ks + (size_t)(col0 + lo) * b_ns];
    acc = __builtin_amdgcn_wmma_f32_16x16x4_f32(
        /*neg_a=*/false, a, /*neg_b=*/false, b,
        /*c_mod=*/(short)0, acc, /*reuse_a=*/false, /*reuse_b=*/false);
  }

#pragma unroll
  for (int r = 0; r < 8; ++r) {
    const int row = row0 + r + 8 * hi;
    const size_t idx = (size_t)row * ldc + col0 + lo;
    if (accumulate) C[idx] += acc[r];
    else            C[idx]  = acc[r];
  }
}

// ---------------------------------------------------------------------------
// Bandwidth-critical kernel: per (batch, chunk, branch), stream CHUNK_L rows
// of mem[b, l, 0:128], compute s_l = row . V[b,:], and accumulate the online
// softmax state (m, Z, acc[128]).  Block = 256 threads = 8 waves; each wave
// streams 4 consecutive rows per iteration (2KB contiguous, one v4f NT load
// per row per lane), giving 64B/lane of independent loads in flight per
// iteration before the serial online-softmax update.
// ---------------------------------------------------------------------------
__global__ __launch_bounds__(256)
void softmax_pass_partial(const float* __restrict__ wm, const float* __restrict__ cm,
                          const float* __restrict__ Vw, const float* __restrict__ Vc,
                          float* __restrict__ Pm, float* __restrict__ Pz,
                          float* __restrict__ Pacc) {
  const int b      = blockIdx.x;
  const int chunk  = blockIdx.y;
  const int branch = blockIdx.z;
  const float* mem = branch == 0 ? wm : cm;
  const float* V   = branch == 0 ? Vw : Vc;

  const int tid  = threadIdx.x;
  const int lane = tid & 31;
  const int wave = tid >> 5;             // 0..7

  const v4f* memv = reinterpret_cast<const v4f*>(mem + (size_t)b * L_SZ * D_SZ);
  const v4f  v    = reinterpret_cast<const v4f*>(V + (size_t)b * D_SZ)[lane];

  float m = -INFINITY;
  float Z = 0.0f;
  v4f   acc = {0.f, 0.f, 0.f, 0.f};

  const int rowBase = chunk * CHUNK_L;
  // 32 rows per block iteration: wave w takes rows rowBase + 32*i + 4*w .. +3
  for (int i = 0; i < CHUNK_L / 32; ++i) {
    const int r0 = rowBase + 32 * i + 4 * wave;
    const v4f* p = memv + (size_t)r0 * 32 + lane;

    // Prefetch 64 rows (32KB) ahead; speculative, dropped if OOB at the tail.
    __builtin_prefetch((const void*)(p + 64 * 32), 0, 0);

    const v4f x0 = __builtin_nontemporal_load(p);
    const v4f x1 = __builtin_nontemporal_load(p + 32);
    const v4f x2 = __builtin_nontemporal_load(p + 64);
    const v4f x3 = __builtin_nontemporal_load(p + 96);

    float s0 = dot4(x0, v);
    float s1 = dot4(x1, v);
    float s2 = dot4(x2, v);
    float s3 = dot4(x3, v);
#pragma unroll
    for (int off = 16; off > 0; off >>= 1) {
      s0 += __shfl_xor(s0, off, 32);
      s1 += __shfl_xor(s1, off, 32);
      s2 += __shfl_xor(s2, off, 32);
      s3 += __shfl_xor(s3, off, 32);
    }

    const float mnew = fmaxf(m, fmaxf(fmaxf(s0, s1), fmaxf(s2, s3)));
    const float corr = __expf(m - mnew);          // 0 on first iter (m=-inf)
    const float p0   = __expf(s0 - mnew);
    const float p1   = __expf(s1 - mnew);
    const float p2   = __expf(s2 - mnew);
    const float p3   = __expf(s3 - mnew);
    Z   = Z * corr + (p0 + p1) + (p2 + p3);
    acc = acc * corr + p0 * x0 + p1 * x1 + p2 * x2 + p3 * x3;
    m   = mnew;
  }

  // Cross-wave merge through LDS, then emit one partial per chunk.
  __shared__ float sm[8];
  __shared__ float sZ[8];
  __shared__ float sacc[8][D_SZ];
  if (lane == 0) { sm[wave] = m; sZ[wave] = Z; }
  sacc[wave][lane * 4 + 0] = acc.x;
  sacc[wave][lane * 4 + 1] = acc.y;
  sacc[wave][lane * 4 + 2] = acc.z;
  sacc[wave][lane * 4 + 3] = acc.w;
  __syncthreads();

  const int part = (branch * B_SZ + b) * NCHUNK + chunk;
  if (tid < D_SZ) {
    float M = sm[0];
#pragma unroll
    for (int w = 1; w < 8; ++w) M = fmaxf(M, sm[w]);
    float Zt = 0.f, A = 0.f;
#pragma unroll
    for (int w = 0; w < 8; ++w) {
      const float f = __expf(sm[w] - M);
      Zt += sZ[w] * f;
      A  += sacc[w][tid] * f;
    }
    Pacc[(size_t)part * D_SZ + tid] = A;
    if (tid == 0) { Pm[part] = M; Pz[part] = Zt; }
  }
}

// Merge the NCHUNK partials per (branch, b) into T[b, 0:128].
__global__ __launch_bounds__(D_SZ)
void softmax_combine(const float* __restrict__ Pm, const float* __restrict__ Pz,
                     const float* __restrict__ Pacc,
                     float* __restrict__ Tw, float* __restrict__ Tc) {
  const int b      = blockIdx.x;
  const int branch = blockIdx.y;
  const int d      = threadIdx.x;       // 0..127
  float* T = branch == 0 ? Tw : Tc;

  const int base = (branch * B_SZ + b) * NCHUNK;
  float M = Pm[base];
#pragma unroll
  for (int c = 1; c < NCHUNK; ++c) M = fmaxf(M, Pm[base + c]);
  float Zt = 0.f, A = 0.f;
#pragma unroll
  for (int c = 0; c < NCHUNK; ++c) {
    const float f = __expf(Pm[base + c] - M);
    Zt += Pz[base + c] * f;
    A  += Pacc[(size_t)(base + c) * D_SZ + d] * f;
  }
  T[(size_t)b * D_SZ + d] = A / Zt;
}

// ---------------------------------------------------------------------------
// Per-hop epilogue: gate = sigmoid(q@G + bias) (q@G precomputed), mix, and
// cross-swap the two branches into the next-hop query buffers.
// ---------------------------------------------------------------------------
__global__ __launch_bounds__(256)
void gate_combine(const float* __restrict__ ctx_w, const float* __restrict__ gl_w,
                  const float* __restrict__ b_w,   const float* __restrict__ q_w,
                  const float* __restrict__ ctx_c, const float* __restrict__ gl_c,
                  const float* __restrict__ b_c,   const float* __restrict__ q_c,
                  float* __restrict__ qw_next, float* __restrict__ qc_next, int n) {
  const int i = blockIdx.x * blockDim.x + threadIdx.x;
  if (i >= n) return;
  const int e = i & (E_SZ - 1);

  const float gw = 1.0f / (1.0f + __expf(-(gl_w[i] + b_w[e])));
  const float nw = ctx_w[i] * gw + q_w[i] * (1.0f - gw);
  const float gc = 1.0f / (1.0f + __expf(-(gl_c[i] + b_c[e])));
  const float nc = ctx_c[i] * gc + q_c[i] * (1.0f - gc);

  qw_next[i] = nc;   // cross swap: wm_in <- cm_dout
  qc_next[i] = nw;   // cm_in <- wm_dout
}

// ---------------------------------------------------------------------------
extern "C" void kernel_launch(void* const* d_in, const int* in_sizes, int n_in,
                              void* d_out, int out_size, void* d_ws, size_t ws_size,
                              hipStream_t stream) {
  const float* wm_input     = (const float*)d_in[0];
  const float* cm_input     = (const float*)d_in[1];
  const float* wm_out_query = (const float*)d_in[2];
  const float* cm_out_query = (const float*)d_in[3];
  const float* E_W = (const float*)d_in[4];
  const float* F_W = (const float*)d_in[5];
  const float* E_C = (const float*)d_in[6];
  const float* F_C = (const float*)d_in[7];
  const float* G_W = (const float*)d_in[8];
  const float* G_C = (const float*)d_in[9];
  const float* b_W = (const float*)d_in[10];
  const float* b_C = (const float*)d_in[11];
  const float* U_W = (const float*)d_in[12];
  const float* U_C = (const float*)d_in[13];

  float* ws = (float*)d_ws;
  float* Vw   = ws;                     // [256,128]
  float* Vc   = Vw   + B_SZ * D_SZ;
  float* Tw   = Vc   + B_SZ * D_SZ;     // [256,128]
  float* Tc   = Tw   + B_SZ * D_SZ;
  float* ctxw = Tc   + B_SZ * D_SZ;     // [256,64]
  float* ctxc = ctxw + B_SZ * E_SZ;
  float* glw  = ctxc + B_SZ * E_SZ;     // [256,64]
  float* glc  = glw  + B_SZ * E_SZ;
  float* qw1  = glc  + B_SZ * E_SZ;     // ping-pong next-hop queries
  float* qc1  = qw1  + B_SZ * E_SZ;
  float* qw2  = qc1  + B_SZ * E_SZ;
  float* qc2  = qw2  + B_SZ * E_SZ;
  float* Pm   = qc2  + B_SZ * E_SZ;     // [2*256*NCHUNK]
  float* Pz   = Pm   + 2 * B_SZ * NCHUNK;
  float* Pacc = Pz   + 2 * B_SZ * NCHUNK; // [2*256*NCHUNK, 128]
  (void)in_sizes; (void)n_in; (void)out_size; (void)ws_size;

  // Cross-wired initial queries.
  const float* qw = cm_out_query;   // waveform branch queried by clinical query
  const float* qc = wm_out_query;

  const dim3 blk32(32);
  const dim3 gV(D_SZ / 16, B_SZ / 16);   // N=128 tiles x M=256 tiles
  const dim3 gE(E_SZ / 16, B_SZ / 16);   // N=64  tiles x M=256 tiles

  for (int hop = 0; hop < 2; ++hop) {
    // V = q @ E^T : K=64, N=128.  B(k=e, n=d) = E[d*64+e] -> strides (1, 64).
    wmma_gemm_f32<<<gV, blk32, 0, stream>>>(qw, E_SZ, E_W, 1, E_SZ, Vw, D_SZ, E_SZ, 0);
    wmma_gemm_f32<<<gV, blk32, 0, stream>>>(qc, E_SZ, E_C, 1, E_SZ, Vc, D_SZ, E_SZ, 0);

    // Fused online-softmax stream over both 512MiB memories (split-L).
    softmax_pass_partial<<<dim3(B_SZ, NCHUNK, 2), 256, 0, stream>>>(
        wm_input, cm_input, Vw, Vc, Pm, Pz, Pacc);
    softmax_combine<<<dim3(B_SZ, 2), D_SZ, 0, stream>>>(Pm, Pz, Pacc, Tw, Tc);

    // ctx = T @ F : K=128, N=64.  F row-major [128,64] -> strides (64, 1).
    wmma_gemm_f32<<<gE, blk32, 0, stream>>>(Tw, D_SZ, F_W, E_SZ, 1, ctxw, E_SZ, D_SZ, 0);
    wmma_gemm_f32<<<gE, blk32, 0, stream>>>(Tc, D_SZ, F_C, E_SZ, 1, ctxc, E_SZ, D_SZ, 0);

    // gate_lin = q @ G : K=64, N=64.
    wmma_gemm_f32<<<gE, blk32, 0, stream>>>(qw, E_SZ, G_W, E_SZ, 1, glw, E_SZ, E_SZ, 0);
    wmma_gemm_f32<<<gE, blk32, 0, stream>>>(qc, E_SZ, G_C, E_SZ, 1, glc, E_SZ, E_SZ, 0);

    float* qwn = (hop == 0) ? qw1 : qw2;
    float* qcn = (hop == 0) ? qc1 : qc2;
    const int n = B_SZ * E_SZ;
    gate_combine<<<(n + 255) / 256, 256, 0, stream>>>(
        ctxw, glw, b_W, qw, ctxc, glc, b_C, qc, qwn, qcn, n);
    qw = qwn;
    qc = qcn;
  }

  // out = qw @ U_W + qc @ U_C   ([256,64] f32)
  float* out = (float*)d_out;
  wmma_gemm_f32<<<gE, blk32, 0, stream>>>(qw, E_SZ, U_W, E_SZ, 1, out, E_SZ, E_SZ, 0);
  wmma_gemm_f32<<<gE, blk32, 0, stream>>>(qc, E_SZ, U_C, E_SZ, 1, out, E_SZ, E_SZ, 1);
}